// KalmanFilter_54597624266755
// MI455X (gfx1250) — compile-verified
//
#include <hip/hip_runtime.h>

#define Bsz 2048
#define Tsz 512
#define Msz 2
#define Ssz 8

typedef float v2f __attribute__((ext_vector_type(2)));
typedef float v4f __attribute__((ext_vector_type(4)));
typedef float v8f __attribute__((ext_vector_type(8)));

// ---------------------------------------------------------------------------
// Kernel A: batch-independent Riccati recursion (sequential in t, one wave).
// Produces per step:
//   wsG[t]  : 16x8 row-major, rows 0..7 = A_t = (I - K_t H) F, rows 8..9 = C=H F,
//             rows 10..15 = 0   (A-operand source for WMMA)
//   wsK[t]  : 16x4 row-major, K_t padded with zeros (A-operand source)
//   wsS[t]  : 2x2 innovation covariance S_t = H P_p H^T + R
// ---------------------------------------------------------------------------
__global__ __launch_bounds__(32) void kf_covar(
    const float* __restrict__ F, const float* __restrict__ H,
    const float* __restrict__ Q, const float* __restrict__ R,
    const float* __restrict__ P0,
    float* __restrict__ wsG, float* __restrict__ wsK, float* __restrict__ wsS)
{
  __shared__ float sF[64], sQ[64], sH[16], sR[4];
  __shared__ float sP[64], sFP[64], sPp[64], sHP[16], sK[16], sSin[4], sBm[64], sC[16];
  const int l = threadIdx.x;
  sF[2*l] = F[2*l];   sF[2*l+1] = F[2*l+1];
  sQ[2*l] = Q[2*l];   sQ[2*l+1] = Q[2*l+1];
  sP[2*l] = P0[2*l];  sP[2*l+1] = P0[2*l+1];
  if (l < 16) sH[l] = H[l];
  if (l < 4)  sR[l] = R[l];
  __syncthreads();
  if (l < 16) {                       // C = H F  (2x8)
    int m = l >> 3, j = l & 7;
    float a = 0.f;
#pragma unroll
    for (int k = 0; k < 8; ++k) a += sH[m*8+k] * sF[k*8+j];
    sC[l] = a;
  }
  __syncthreads();
  const int i  = l >> 2;              // this lane owns row i, cols c0..c0+1
  const int c0 = (l & 3) * 2;
  for (int t = 0; t < Tsz; ++t) {
    // FP = F * P
#pragma unroll
    for (int d = 0; d < 2; ++d) {
      int k = c0 + d; float a = 0.f;
#pragma unroll
      for (int j = 0; j < 8; ++j) a += sF[i*8+j] * sP[j*8+k];
      sFP[i*8+k] = a;
    }
    __syncthreads();
    // Pp = FP * F^T + Q
#pragma unroll
    for (int d = 0; d < 2; ++d) {
      int k = c0 + d; float a = sQ[i*8+k];
#pragma unroll
      for (int j = 0; j < 8; ++j) a += sFP[i*8+j] * sF[k*8+j];
      sPp[i*8+k] = a;
    }
    __syncthreads();
    // HP = H * Pp   (2x8)
    if (l < 16) {
      int m = l >> 3, k = l & 7; float a = 0.f;
#pragma unroll
      for (int j = 0; j < 8; ++j) a += sH[m*8+j] * sPp[j*8+k];
      sHP[m*8+k] = a;
    }
    __syncthreads();
    // S = HP * H^T + R ; 2x2 inverse
    if (l == 0) {
      float s00 = sR[0], s01 = sR[1], s10 = sR[2], s11 = sR[3];
#pragma unroll
      for (int k = 0; k < 8; ++k) {
        s00 += sHP[k]   * sH[k];
        s01 += sHP[k]   * sH[8+k];
        s10 += sHP[8+k] * sH[k];
        s11 += sHP[8+k] * sH[8+k];
      }
      wsS[t*4+0] = s00; wsS[t*4+1] = s01; wsS[t*4+2] = s10; wsS[t*4+3] = s11;
      float inv = 1.f / (s00*s11 - s01*s10);
      sSin[0] =  s11*inv; sSin[1] = -s01*inv; sSin[2] = -s10*inv; sSin[3] = s00*inv;
    }
    __syncthreads();
    // K[s,m] = sum_n Sinv[m,n] * HP[n,s]   (K = solve(S,HP)^T, 8x2)
    if (l < 16) {
      int s = l >> 1, m = l & 1;
      sK[s*2+m] = sSin[m*2+0]*sHP[0*8+s] + sSin[m*2+1]*sHP[1*8+s];
    }
    __syncthreads();
    // Bm = I - K H
#pragma unroll
    for (int d = 0; d < 2; ++d) {
      int j = c0 + d;
      sBm[i*8+j] = ((i==j)?1.f:0.f) - (sK[i*2+0]*sH[j] + sK[i*2+1]*sH[8+j]);
    }
    __syncthreads();
    // A_t = Bm * F  -> G rows 0..7 ; G rows 8..15 = [C ; 0] ; Kpad ; P_new
#pragma unroll
    for (int d = 0; d < 2; ++d) {
      int j = c0 + d; float a = 0.f;
#pragma unroll
      for (int k = 0; k < 8; ++k) a += sBm[i*8+k] * sF[k*8+j];
      wsG[t*128 + i*8 + j] = a;
    }
#pragma unroll
    for (int d = 0; d < 2; ++d) {
      int e = 64 + 2*l + d, gi = e >> 3, gj = e & 7;
      wsG[t*128 + e] = (gi < 10) ? sC[(gi-8)*8 + gj] : 0.f;
    }
#pragma unroll
    for (int d = 0; d < 2; ++d) {
      int e = 2*l + d, r = e >> 2, c = e & 3;
      wsK[t*64 + e] = (r < 8 && c < 2) ? sK[r*2 + c] : 0.f;
    }
    // P_new = Pp - K * HP
    float pn0, pn1;
    {
      int k = c0;
      pn0 = sPp[i*8+k]   - (sK[i*2+0]*sHP[k]   + sK[i*2+1]*sHP[8+k]);
      pn1 = sPp[i*8+k+1] - (sK[i*2+0]*sHP[k+1] + sK[i*2+1]*sHP[8+k+1]);
    }
    __syncthreads();
    sP[i*8+c0] = pn0; sP[i*8+c0+1] = pn1;
    __syncthreads();
  }
}

// ---------------------------------------------------------------------------
// Kernel B: per-batch affine recurrence via V_WMMA_F32_16X16X4_F32.
// One wave = 16 batches. State kept as D fragment: D[s,b] = m[b][s],
// lane b holds VGPR s (s<8).  Per step:
//   acc  = Kpad(16x4) x Y^T(4x16)                       (K y term)
//   acc += Gc0(16x4)  x state^T rows 0..3
//   acc += Gc1(16x4)  x state^T rows 4..7
// Rows 8..9 of acc (upper lanes, VGPR 0..1) = H F m_{t-1} = pred_means[.,t,:].
// ---------------------------------------------------------------------------
__global__ __launch_bounds__(32) void kf_mean(
    const float* __restrict__ y, const float* __restrict__ m0,
    const float* __restrict__ wsG, const float* __restrict__ wsK,
    float* __restrict__ pm)
{
  const int l    = threadIdx.x;
  const int r    = l & 15;        // A-op row / batch-within-block
  const int half = l >> 4;        // 0: K cols {0,1} / 1: K cols {2,3}
  const int base = half * 2;
  const int batch = blockIdx.x * 16 + r;
  const float* gp = wsG + r*8 + base;                 // A-op gather for G
  const float* kp = wsK + r*4 + base;                 // A-op gather for Kpad
  const float* yp = y  + (size_t)batch * (Tsz*Msz);
  float*       po = pm + (size_t)batch * (Tsz*Msz);

  v8f st;
#pragma unroll
  for (int s = 0; s < 8; ++s) st[s] = m0[s];          // broadcast initial mean

  for (int t = 0; t < Tsz; ++t) {
    v2f gc0 = *(const v2f*)(gp + t*128);              // G cols base..base+1
    v2f gc1 = *(const v2f*)(gp + t*128 + 4);          // G cols 4+base..5+base
    v2f kc  = *(const v2f*)(kp + t*64);               // Kpad cols base..base+1
    v2f yv  = *(const v2f*)(yp + t*2);
    v2f yb; yb[0] = half ? 0.f : yv[0]; yb[1] = half ? 0.f : yv[1];  // Y^T B-op

    v8f acc = {0.f,0.f,0.f,0.f,0.f,0.f,0.f,0.f};
    acc = __builtin_amdgcn_wmma_f32_16x16x4_f32(false, kc, false, yb,
                                                (short)0, acc, false, false);
    // D-layout state -> B-operand (4x16) chunks: upper half-wave needs rows 2,3 / 6,7
    float u2 = __shfl_xor(st[2], 16, 32);
    float u3 = __shfl_xor(st[3], 16, 32);
    float u6 = __shfl_xor(st[6], 16, 32);
    float u7 = __shfl_xor(st[7], 16, 32);
    v2f b0; b0[0] = half ? u2 : st[0]; b0[1] = half ? u3 : st[1];
    v2f b1; b1[0] = half ? u6 : st[4]; b1[1] = half ? u7 : st[5];

    acc = __builtin_amdgcn_wmma_f32_16x16x4_f32(false, gc0, false, b0,
                                                (short)0, acc, false, false);
    acc = __builtin_amdgcn_wmma_f32_16x16x4_f32(false, gc1, false, b1,
                                                (short)0, acc, false, false);
    if (half) {                                        // rows 8,9 = Hm for batch r
      v2f hm; hm[0] = acc[0]; hm[1] = acc[1];
      *(v2f*)(po + t*2) = hm;
    }
    st = acc;                                          // rows 0..7 = new posterior
  }
}

// ---------------------------------------------------------------------------
// Kernel C: pred_covs[b,t,:,:] = S_t  (pure broadcast, b128 stores)
// ---------------------------------------------------------------------------
__global__ void kf_covout(const float* __restrict__ wsS, float* __restrict__ pc)
{
  size_t idx = (size_t)blockIdx.x * blockDim.x + threadIdx.x;   // over B*T
  if (idx >= (size_t)Bsz * Tsz) return;
  int t = (int)(idx % Tsz);
  v4f s = *(const v4f*)(wsS + t*4);
  *(v4f*)(pc + idx*4) = s;
}

extern "C" void kernel_launch(void* const* d_in, const int* in_sizes, int n_in,
                              void* d_out, int out_size, void* d_ws, size_t ws_size,
                              hipStream_t stream) {
  const float* y  = (const float*)d_in[0];
  const float* F  = (const float*)d_in[1];
  const float* H  = (const float*)d_in[2];
  const float* Q  = (const float*)d_in[3];
  const float* R  = (const float*)d_in[4];
  const float* m0 = (const float*)d_in[5];
  const float* P0 = (const float*)d_in[6];

  float* pm = (float*)d_out;                       // [B,T,M]
  float* pc = pm + (size_t)Bsz * Tsz * Msz;        // [B,T,M,M]

  float* ws  = (float*)d_ws;
  float* wsG = ws;                 // 512 * 128 floats
  float* wsK = wsG + Tsz * 128;    // 512 *  64 floats
  float* wsS = wsK + Tsz * 64;     // 512 *   4 floats

  kf_covar<<<1, 32, 0, stream>>>(F, H, Q, R, P0, wsG, wsK, wsS);
  kf_mean<<<Bsz / 16, 32, 0, stream>>>(y, m0, wsG, wsK, pm);
  const int nth = 256;
  const int nbl = (int)(((size_t)Bsz * Tsz + nth - 1) / nth);
  kf_covout<<<nbl, nth, 0, stream>>>(wsS, pc);
}